// ComposeTransform_5600637354404
// MI455X (gfx1250) — compile-verified
//
#include <hip/hip_runtime.h>

// Compose of three dense displacement fields (voxelmorph-style), fully fused:
//   curr1 = t3 + warp(t2, t3);  out = curr1 + warp(t1, curr1)
// Field layout: [B=2, D=128, H=160, W=192, C=3], C innermost, f32.
//
// Grid mapping (no div/mod anywhere):
//   threadIdx.x = w (0..191), blockIdx.x = h, blockIdx.y = d, blockIdx.z = b
// All element offsets are signed int (max ~23.6M < 2^31) so the backend can
// assume no-wrap and merge each 3-float channel access into global_load_b96.

#define DD 128
#define HH 160
#define WW 192
#define VOX (DD * HH * WW)          // voxels per batch
#define NBATCH 2

// Trilinear interpolation of vol [D,H,W,3] at clipped absolute coords,
// matching neurite interpn: clip to [0, max], floor/ceil corners,
// weight(floor) = ceil - p, weight(ceil) = 1 - weight(floor).
__device__ __forceinline__ void trilerp3(const float* __restrict__ vol,
                                         float pd, float ph, float pw,
                                         float r[3]) {
    const float maxd = (float)(DD - 1);
    const float maxh = (float)(HH - 1);
    const float maxw = (float)(WW - 1);

    pd = fminf(fmaxf(pd, 0.0f), maxd);
    ph = fminf(fmaxf(ph, 0.0f), maxh);
    pw = fminf(fmaxf(pw, 0.0f), maxw);

    float fd = floorf(pd), fh = floorf(ph), fw = floorf(pw);
    float cd = fminf(fd + 1.0f, maxd);
    float ch = fminf(fh + 1.0f, maxh);
    float cw = fminf(fw + 1.0f, maxw);

    int id0 = (int)fd, ih0 = (int)fh, iw0 = (int)fw;
    int id1 = (int)cd, ih1 = (int)ch, iw1 = (int)cw;

    float wdf = cd - pd, wdc = 1.0f - wdf;   // floor / ceil weights, dim d
    float whf = ch - ph, whc = 1.0f - whf;   // dim h
    float wwf = cw - pw, wwc = 1.0f - wwf;   // dim w

    float a0 = 0.0f, a1 = 0.0f, a2 = 0.0f;
#pragma unroll
    for (int ad = 0; ad < 2; ++ad) {
        const int   dI = ad ? id1 : id0;
        const float dW = ad ? wdc : wdf;
#pragma unroll
        for (int ah = 0; ah < 2; ++ah) {
            const int   hI = ah ? ih1 : ih0;
            const float hW = dW * (ah ? whc : whf);
            const int rowBase = (dI * HH + hI) * WW;
#pragma unroll
            for (int aw = 0; aw < 2; ++aw) {
                const int   wI = aw ? iw1 : iw0;
                const float wt = hW * (aw ? wwc : wwf);
                // Signed-int addressing -> no-wrap assumption -> one b96 load.
                const float* __restrict__ p = vol + 3 * (rowBase + wI);
                a0 = fmaf(wt, p[0], a0);
                a1 = fmaf(wt, p[1], a1);
                a2 = fmaf(wt, p[2], a2);
            }
        }
    }
    r[0] = a0; r[1] = a1; r[2] = a2;
}

__global__ __launch_bounds__(WW)
void compose3_fused_kernel(const float* __restrict__ t1,
                           const float* __restrict__ t2,
                           const float* __restrict__ t3,
                           float* __restrict__ out) {
    const int w = threadIdx.x;   // 0..WW-1 (contiguous -> coalesced b96)
    const int h = blockIdx.x;    // 0..HH-1
    const int d = blockIdx.y;    // 0..DD-1
    const int b = blockIdx.z;    // 0..NBATCH-1

    const int vox   = (d * HH + h) * WW + w;
    const int vbase = b * (VOX * 3);            // uniform (SALU)
    const int o     = vbase + vox * 3;

    const float* __restrict__ T1 = t1 + vbase;
    const float* __restrict__ T2 = t2 + vbase;

    // t3 is streamed exactly once: non-temporal load (TH=NT) keeps L2
    // capacity for the t2/t1 gather working sets (94 MB each, L2 = 192 MB).
    const float s0 = __builtin_nontemporal_load(t3 + o + 0);
    const float s1 = __builtin_nontemporal_load(t3 + o + 1);
    const float s2 = __builtin_nontemporal_load(t3 + o + 2);

    // Stage 1: curr1 = t3 + trilerp(t2, grid + t3)
    float a[3];
    trilerp3(T2, (float)d + s0, (float)h + s1, (float)w + s2, a);
    const float c0 = s0 + a[0];
    const float c1 = s1 + a[1];
    const float c2 = s2 + a[2];

    // Stage 2: out = curr1 + trilerp(t1, grid + curr1)
    trilerp3(T1, (float)d + c0, (float)h + c1, (float)w + c2, a);

    // Output is streamed exactly once: non-temporal stores bypass-rinse L2.
    __builtin_nontemporal_store(c0 + a[0], out + o + 0);
    __builtin_nontemporal_store(c1 + a[1], out + o + 1);
    __builtin_nontemporal_store(c2 + a[2], out + o + 2);
}

extern "C" void kernel_launch(void* const* d_in, const int* in_sizes, int n_in,
                              void* d_out, int out_size, void* d_ws, size_t ws_size,
                              hipStream_t stream) {
    (void)in_sizes; (void)n_in; (void)out_size; (void)d_ws; (void)ws_size;
    const float* t1 = (const float*)d_in[0];
    const float* t2 = (const float*)d_in[1];
    const float* t3 = (const float*)d_in[2];
    float* out = (float*)d_out;

    dim3 block(WW, 1, 1);             // 192 threads = 6 wave32, one row per block
    dim3 grid(HH, DD, NBATCH);        // 160 x 128 x 2 blocks, no index division
    compose3_fused_kernel<<<grid, block, 0, stream>>>(t1, t2, t3, out);
}